// SelfAttention_57870389346662
// MI455X (gfx1250) — compile-verified
//
#include <hip/hip_runtime.h>
#include <stdint.h>

#define B_  2
#define H_  16
#define S_  2048
#define D_  64
#define NQT (S_/16)   // 128 query tiles per (b,h)

typedef float        v2f __attribute__((ext_vector_type(2)));
typedef float        v4f __attribute__((ext_vector_type(4)));
typedef float        v8f __attribute__((ext_vector_type(8)));
typedef unsigned int v4u __attribute__((ext_vector_type(4)));
typedef int          v4i __attribute__((ext_vector_type(4)));
typedef int          v8i __attribute__((ext_vector_type(8)));

#if __has_builtin(__builtin_amdgcn_tensor_load_to_lds) && \
    __has_builtin(__builtin_amdgcn_tensor_store_from_lds) && \
    __has_builtin(__builtin_amdgcn_s_wait_tensorcnt)
#define USE_TDM 1
#else
#define USE_TDM 0
#endif

#if USE_TDM
// ---------------------------------------------------------------------------
// 2-D TDM transfer (load: global->LDS, store: LDS->global). D# built per
// CDNA5 ISA: count=1, type=2, data_size=4B.
// Tracked by TENSORcnt; loads+stores from one wave complete in issue order.
// ---------------------------------------------------------------------------
__device__ __forceinline__ void tdm_2d(const void* ldsp, const void* gptr,
                                       uint32_t tile_d0, uint32_t tile_d1,
                                       uint32_t stride_d0,
                                       uint32_t tensor_d0, uint32_t tensor_d1,
                                       bool store) {
  uint32_t lds_off = (uint32_t)(uintptr_t)ldsp;   // low 32 bits of flat = LDS byte addr
  uint64_t ga = (uint64_t)(uintptr_t)gptr;
  v4u g0;
  g0.x = 1u;                                               // count=1, is_restore=0
  g0.y = lds_off;                                          // lds_addr (bits 63:32)
  g0.z = (uint32_t)(ga & 0xFFFFFFFFu);                     // global_addr[31:0]
  g0.w = (uint32_t)((ga >> 32) & 0x01FFFFFFu) | (2u << 30);// global_addr[56:32], type=2
  v8i g1;
  g1[0] = (int)(2u << 16);                                 // wg_mask=0, data_size=2 -> 4B
  g1[1] = (int)((tensor_d0 & 0xFFFFu) << 16);              // tensor_dim0 lo16
  g1[2] = (int)(((tensor_d0 >> 16) & 0xFFFFu) | ((tensor_d1 & 0xFFFFu) << 16));
  g1[3] = (int)(((tensor_d1 >> 16) & 0xFFFFu) | ((tile_d0 & 0xFFFFu) << 16));
  g1[4] = (int)(tile_d1 & 0xFFFFu);                        // tile_dim1, tile_dim2=0
  g1[5] = (int)stride_d0;                                  // tensor_dim0_stride low32
  g1[6] = 0;
  g1[7] = 0;
  v4i z4 = {0, 0, 0, 0};
#if defined(__clang_major__) && (__clang_major__ >= 23)
  v8i z8 = {0, 0, 0, 0, 0, 0, 0, 0};
  if (store) __builtin_amdgcn_tensor_store_from_lds(g0, g1, z4, z4, z8, 0);
  else       __builtin_amdgcn_tensor_load_to_lds(g0, g1, z4, z4, z8, 0);
#else
  if (store) __builtin_amdgcn_tensor_store_from_lds(g0, g1, z4, z4, 0);
  else       __builtin_amdgcn_tensor_load_to_lds(g0, g1, z4, z4, 0);
#endif
}
#define TWAIT(n) do { __builtin_amdgcn_s_wait_tensorcnt(n); \
                      asm volatile("" ::: "memory"); } while (0)
#else
#define TWAIT(n) asm volatile("s_wait_dscnt 0" ::: "memory")
#endif

// 16x64 fp32 K/V/Q tile (contiguous rows, stride D_) -> LDS
__device__ __forceinline__ void tile_load_16xD(float* ldsp, const float* gsrc) {
#if USE_TDM
  tdm_2d(ldsp, gsrc, /*tile*/ D_, 16, /*stride*/ D_, /*tensor*/ D_, S_, false);
#else
  const int lane = (int)(threadIdx.x & 31u);
  #pragma unroll
  for (int i = lane * 4; i < 16 * D_; i += 32 * 4)
    *(v4f*)&ldsp[i] = *(const v4f*)&gsrc[i];
#endif
}

__device__ __forceinline__ v8f wmma_f32(v2f a, v2f b, v8f c) {
  // V_WMMA_F32_16X16X4_F32: D = A(16x4) x B(4x16) + C, full fp32
  return __builtin_amdgcn_wmma_f32_16x16x4_f32(false, a, false, b, (short)0, c,
                                               false, false);
}

__device__ __forceinline__ float rowmax16(float x) {
  #pragma unroll
  for (int off = 8; off; off >>= 1) x = fmaxf(x, __shfl_xor(x, off, 32));
  return x;
}
__device__ __forceinline__ float rowsum16(float x) {
  #pragma unroll
  for (int off = 8; off; off >>= 1) x += __shfl_xor(x, off, 32);
  return x;
}

// ---------------------------------------------------------------------------
// One wave32 per block; block owns 16 query rows of one (b,h) slice.
// Pass 1: online (m, l) over causal key tiles (double-buffered TDM K loads).
// Pass 2: recompute scores, normalized P tile -> LDS, TDM 2-D strided store
// of the attention weights, O = P*V via WMMA. All tile DMA double-buffered.
// ---------------------------------------------------------------------------
__global__ __launch_bounds__(32) void attn_f32_wmma_kernel(
    const float* __restrict__ q, const float* __restrict__ k,
    const float* __restrict__ v, float* __restrict__ out,
    float* __restrict__ attn) {
  __shared__ float Qs[16 * D_];
  __shared__ float Ks[2][16 * D_];
  __shared__ float Vs[2][16 * D_];
  __shared__ float Ps[2][16 * 16];

  const int lane = (int)(threadIdx.x & 31u);
  const int half = lane >> 4;              // lane group: 0 or 1
  const int l16  = lane & 15;
  const int bh   = (int)blockIdx.x / NQT;  // 0..B*H-1
  const int qt   = (int)blockIdx.x % NQT;  // query tile index
  const int qbase = qt * 16;
  const size_t base = (size_t)bh * S_ * D_;
  float* const attn_bh = attn + (size_t)bh * S_ * S_;
  const float scale = 0.125f;              // 1/sqrt(64)

  // ---- stage Q tile, build A fragments (row = l16, K = kk*4 + 2*half + v) ----
  tile_load_16xD(Qs, q + base + (size_t)qbase * D_);
  TWAIT(0);
  v2f qf[16];
  #pragma unroll
  for (int kk = 0; kk < 16; ++kk) {
    qf[kk].x = Qs[l16 * D_ + kk * 4 + half * 2 + 0];
    qf[kk].y = Qs[l16 * D_ + kk * 4 + half * 2 + 1];
  }

  // ---- pass 1: per-row running max m and denom l (K loads double-buffered) ----
  float m[8], lsum[8];
  #pragma unroll
  for (int r = 0; r < 8; ++r) { m[r] = -__builtin_inff(); lsum[r] = 0.f; }

  tile_load_16xD(Ks[0], k + base);
  for (int kt = 0; kt <= qt; ++kt) {
    const bool pre = (kt < qt);
    if (pre) tile_load_16xD(Ks[(kt + 1) & 1], k + base + (size_t)(kt + 1) * 16 * D_);
    if (pre) TWAIT(1); else TWAIT(0);
    const float* Kb = Ks[kt & 1];
    v8f c = {0.f, 0.f, 0.f, 0.f, 0.f, 0.f, 0.f, 0.f};
    #pragma unroll
    for (int kk = 0; kk < 16; ++kk) {
      v2f bf;
      bf.x = Kb[l16 * D_ + kk * 4 + half * 2 + 0];
      bf.y = Kb[l16 * D_ + kk * 4 + half * 2 + 1];
      c = wmma_f32(qf[kk], bf, c);
    }
    #pragma unroll
    for (int r = 0; r < 8; ++r) {
      float s = c[r] * scale;
      const int row = qbase + r + 8 * half;   // C layout: M = r + 8*half
      const int col = kt * 16 + l16;          //           N = l16
      if (col > row) s = -__builtin_inff();   // causal mask
      const float mn = fmaxf(m[r], rowmax16(s));
      const float e  = __expf(s - mn);
      lsum[r] = lsum[r] * __expf(m[r] - mn) + rowsum16(e);
      m[r] = mn;
    }
  }
  float invl[8];
  #pragma unroll
  for (int r = 0; r < 8; ++r) invl[r] = 1.0f / lsum[r];

  // ---- pass 2: weights out (TDM strided store) + O = P*V ----
  v8f oacc[4];
  #pragma unroll
  for (int db = 0; db < 4; ++db)
    oacc[db] = (v8f){0.f, 0.f, 0.f, 0.f, 0.f, 0.f, 0.f, 0.f};

  tile_load_16xD(Ks[0], k + base);
  tile_load_16xD(Vs[0], v + base);
  for (int kt = 0; kt <= qt; ++kt) {
    const bool pre = (kt < qt);
    if (pre) {
      tile_load_16xD(Ks[(kt + 1) & 1], k + base + (size_t)(kt + 1) * 16 * D_);
      tile_load_16xD(Vs[(kt + 1) & 1], v + base + (size_t)(kt + 1) * 16 * D_);
    }
    // outstanding newer than this tile's loads: prefetch pair + previous P store
    if (pre) { if (kt >= 1) TWAIT(3); else TWAIT(2); }
    else     { if (kt >= 1) TWAIT(1); else TWAIT(0); }

    const float* Kb = Ks[kt & 1];
    const float* Vb = Vs[kt & 1];
    float* Pb = Ps[kt & 1];

    v8f c = {0.f, 0.f, 0.f, 0.f, 0.f, 0.f, 0.f, 0.f};
    #pragma unroll
    for (int kk = 0; kk < 16; ++kk) {
      v2f bf;
      bf.x = Kb[l16 * D_ + kk * 4 + half * 2 + 0];
      bf.y = Kb[l16 * D_ + kk * 4 + half * 2 + 1];
      c = wmma_f32(qf[kk], bf, c);
    }
    #pragma unroll
    for (int r = 0; r < 8; ++r) {
      float s = c[r] * scale;
      const int row = qbase + r + 8 * half;
      const int col = kt * 16 + l16;
      if (col > row) s = -__builtin_inff();
      const float w = __expf(s - m[r]) * invl[r];   // exact softmax weight
      Pb[(r + 8 * half) * 16 + l16] = w;            // C-layout -> LDS
#if !USE_TDM
      attn_bh[(size_t)row * S_ + col] = w;
#endif
    }
    asm volatile("s_wait_dscnt 0" ::: "memory");    // P tile visible to TDM + wave
#if USE_TDM
    // async 2-D strided store of the 16x16 weight tile: rows stride S_
    tdm_2d(Pb, attn_bh + (size_t)qbase * S_ + kt * 16,
           /*tile*/ 16, 16, /*stride*/ S_, /*tensor*/ 16, 16, true);
#endif
    #pragma unroll
    for (int kk2 = 0; kk2 < 4; ++kk2) {
      v2f pa;
      pa.x = Pb[l16 * 16 + kk2 * 4 + half * 2 + 0];
      pa.y = Pb[l16 * 16 + kk2 * 4 + half * 2 + 1];
      #pragma unroll
      for (int db = 0; db < 4; ++db) {
        v2f vb;
        vb.x = Vb[(kk2 * 4 + half * 2 + 0) * D_ + db * 16 + l16];
        vb.y = Vb[(kk2 * 4 + half * 2 + 1) * D_ + db * 16 + l16];
        oacc[db] = wmma_f32(pa, vb, oacc[db]);
      }
    }
  }

  // ---- write O tile ----
  #pragma unroll
  for (int db = 0; db < 4; ++db)
    #pragma unroll
    for (int r = 0; r < 8; ++r)
      out[base + (size_t)(qbase + r + 8 * half) * D_ + db * 16 + l16] = oacc[db][r];

  // ---- zero-fill masked (upper triangular) attn region: softmax == 0 there ----
  const int zstart = (qt + 1) * 16;
  for (int row = 0; row < 16; ++row) {
    float* dst = attn_bh + (size_t)(qbase + row) * S_;
    for (int col = zstart + lane * 4; col < S_; col += 32 * 4)
      *(v4f*)&dst[col] = (v4f){0.f, 0.f, 0.f, 0.f};
  }
  // pending TDM store (if any) is flushed by S_ENDPGM's implicit wait-idle
}

extern "C" void kernel_launch(void* const* d_in, const int* in_sizes, int n_in,
                              void* d_out, int out_size, void* d_ws, size_t ws_size,
                              hipStream_t stream) {
  (void)in_sizes; (void)n_in; (void)d_ws; (void)ws_size; (void)out_size;
  const float* q = (const float*)d_in[0];
  const float* k = (const float*)d_in[1];
  const float* v = (const float*)d_in[2];
  // d_in[3] = causal mask, known statically -> unused
  float* out  = (float*)d_out;                              // [B,H,S,D]
  float* attn = out + (size_t)B_ * H_ * S_ * D_;            // [B,H,S,S]
  dim3 grid(B_ * H_ * NQT);
  dim3 block(32);
  attn_f32_wmma_kernel<<<grid, block, 0, stream>>>(q, k, v, out, attn);
}